// MultiBoxLoss_12506944766687
// MI455X (gfx1250) — compile-verified
//
#include <hip/hip_runtime.h>
#include <hip/hip_bf16.h>

// MultiBoxLoss (SSD) for gfx1250.
// N=64 images, P=8732 priors, C=91 classes, M=20 padded objects.
#define N_IMG 64
#define P_PRI 8732
#define C_CLS 91
#define M_OBJ 20
#define SORT_N 16384           // next pow2 >= P_PRI, for bitonic sort
#define TILE_F (16 * C_CLS)    // 1456 floats per wave tile (contiguous, stride 91)

typedef float v2f __attribute__((ext_vector_type(2)));
typedef float v8f __attribute__((ext_vector_type(8)));

__device__ __forceinline__ float smooth_l1(float d) {
    float a = fabsf(d);
    return (a < 1.0f) ? 0.5f * d * d : a - 0.5f;
}

// ---------------------------------------------------------------------------
// Kernel 1: per-image prior<->object matching. One block per image.
// ---------------------------------------------------------------------------
__global__ void mbl_match_kernel(const float* __restrict__ boxes,    // (N,M,4) xyxy
                                 const int*   __restrict__ labels,   // (N,M)
                                 const float* __restrict__ priors,   // (P,4) cxcywh
                                 float* __restrict__ ov_g,           // (N,P) scratch
                                 int*   __restrict__ obj_g,          // (N,P)
                                 int*   __restrict__ tcls)           // (N,P) out labels
{
    const int n   = blockIdx.x;
    const int tid = threadIdx.x;

    __shared__ float sbx[M_OBJ][4];
    __shared__ float sarea[M_OBJ];
    __shared__ int   slab[M_OBJ];
    __shared__ unsigned long long sbest[M_OBJ];   // packed (iou_bits<<32)|(~p)

    if (tid < M_OBJ) {
        float x1 = boxes[(n * M_OBJ + tid) * 4 + 0];
        float y1 = boxes[(n * M_OBJ + tid) * 4 + 1];
        float x2 = boxes[(n * M_OBJ + tid) * 4 + 2];
        float y2 = boxes[(n * M_OBJ + tid) * 4 + 3];
        sbx[tid][0] = x1; sbx[tid][1] = y1; sbx[tid][2] = x2; sbx[tid][3] = y2;
        sarea[tid]  = (x2 - x1) * (y2 - y1);
        slab[tid]   = labels[n * M_OBJ + tid];
        sbest[tid]  = 0xFFFFFFFFull;  // iou=0.0, prior=0
    }
    __syncthreads();

    unsigned long long best[M_OBJ];
#pragma unroll
    for (int m = 0; m < M_OBJ; ++m) best[m] = 0ull;

    for (int p = tid; p < P_PRI; p += blockDim.x) {
        float pcx = priors[p * 4 + 0], pcy = priors[p * 4 + 1];
        float pw  = priors[p * 4 + 2], ph  = priors[p * 4 + 3];
        float bx1 = pcx - pw * 0.5f, by1 = pcy - ph * 0.5f;
        float bx2 = pcx + pw * 0.5f, by2 = pcy + ph * 0.5f;
        float area_b = (bx2 - bx1) * (by2 - by1);

        float best_ov = -1.0f;
        int   best_m  = 0;
#pragma unroll
        for (int m = 0; m < M_OBJ; ++m) {
            float lx = fmaxf(sbx[m][0], bx1), ly = fmaxf(sbx[m][1], by1);
            float hx = fminf(sbx[m][2], bx2), hy = fminf(sbx[m][3], by2);
            float iw = fmaxf(hx - lx, 0.0f),  ih = fmaxf(hy - ly, 0.0f);
            float inter = iw * ih;
            float iou = inter / (sarea[m] + area_b - inter);
            if (iou > best_ov) { best_ov = iou; best_m = m; }   // first-max
            unsigned long long pk =
                ((unsigned long long)__float_as_uint(iou) << 32) |
                (unsigned long long)(0xFFFFFFFFu - (unsigned)p); // ties -> min p
            if (pk > best[m]) best[m] = pk;
        }
        ov_g[n * P_PRI + p]  = best_ov;
        obj_g[n * P_PRI + p] = best_m;
    }
#pragma unroll
    for (int m = 0; m < M_OBJ; ++m) atomicMax(&sbest[m], best[m]);
    __syncthreads();

    if (tid == 0) {   // force each object onto its best prior (last write wins)
        for (int m = 0; m < M_OBJ; ++m) {
            unsigned p = 0xFFFFFFFFu - (unsigned)(sbest[m] & 0xFFFFFFFFull);
            ov_g[n * P_PRI + p]  = 1.0f;
            obj_g[n * P_PRI + p] = m;
        }
    }
    __syncthreads();

    for (int p = tid; p < P_PRI; p += blockDim.x) {
        int   m  = obj_g[n * P_PRI + p];
        float ov = ov_g[n * P_PRI + p];
        tcls[n * P_PRI + p] = (ov < 0.5f) ? 0 : slab[m];
    }
}

// ---------------------------------------------------------------------------
// Kernel 2: per-prior cross entropy (the 203 MB streaming pass).
// Each wave32 handles 16 consecutive priors (tile contiguous in memory).
// Fast path: whole 1456-float tile copied as 364 float4 (b128 VMEM + b128 DS).
// sum-exp accumulated with V_WMMA_F32_16X16X4_F32 vs all-ones B (exact f32).
// ---------------------------------------------------------------------------
__global__ void mbl_ce_kernel(const float* __restrict__ scores,  // (N,P,C)
                              const int*   __restrict__ tcls,    // (N,P)
                              float* __restrict__ ce_g)          // (N,P)
{
    const int n    = blockIdx.y;
    const int wave = threadIdx.x >> 5;
    const int lane = threadIdx.x & 31;
    const int p0   = blockIdx.x * 128 + wave * 16;

    __shared__ float tile[8][TILE_F];
    float* t = tile[wave];

    if (p0 + 16 <= P_PRI) {
        // Interior wave: tile is contiguous & 16B aligned in global memory.
        const float4* src4 = (const float4*)(scores + (size_t)(n * P_PRI + p0) * C_CLS);
        float4* dst4 = (float4*)t;
        for (int idx = lane; idx < TILE_F / 4; idx += 32)   // 364 float4
            dst4[idx] = src4[idx];
    } else {
        // Tail wave: per-element copy with row clamp.
        for (int idx = lane; idx < TILE_F; idx += 32) {
            int r = idx / C_CLS;
            int c = idx - r * C_CLS;
            int p = p0 + r; if (p > P_PRI - 1) p = P_PRI - 1;
            t[r * C_CLS + c] = scores[((size_t)(n * P_PRI + p)) * C_CLS + c];
        }
    }
    __syncthreads();

    const int    m    = lane & 15;       // row of A this lane serves
    const int    half = lane >> 4;       // 0: K0/K1, 1: K2/K3  (ISA A-layout)
    const float* row  = t + m * C_CLS;

    // Row max: half-wave split + xor-16 combine
    float mym = -3.4e38f;
    const int cb   = half ? 46 : 0;
    const int cend = half ? C_CLS : 46;
    for (int c = cb; c < cend; ++c) mym = fmaxf(mym, row[c]);
    const float rmax = fmaxf(mym, __shfl_xor(mym, 16, 32));

    // Accumulate exp(x - rmax) row sums on the matrix unit: D = A * ones + C
    v8f acc = {0.f, 0.f, 0.f, 0.f, 0.f, 0.f, 0.f, 0.f};
    v2f bones; bones.x = 1.0f; bones.y = 1.0f;
    for (int k = 0; k < 92; k += 4) {       // 23 chunks covering K=0..91
        const int c0 = k + half * 2;
        v2f a;
        a.x = (c0     < C_CLS) ? __expf(row[c0]     - rmax) : 0.0f;
        a.y = (c0 + 1 < C_CLS) ? __expf(row[c0 + 1] - rmax) : 0.0f;
        acc = __builtin_amdgcn_wmma_f32_16x16x4_f32(
            /*neg_a=*/false, a, /*neg_b=*/false, bones,
            /*c_mod=*/(short)0, acc, /*reuse_a=*/false, /*reuse_b=*/false);
    }

    // Harvest: D layout -> VGPR r holds row r (lanes 0-15) / row r+8 (16-31)
    const int q = lane & 7;
    float e = (q == 0) ? acc[0] : (q == 1) ? acc[1] : (q == 2) ? acc[2] :
              (q == 3) ? acc[3] : (q == 4) ? acc[4] : (q == 5) ? acc[5] :
              (q == 6) ? acc[6] : acc[7];
    const int src = ((lane & 8) ? (lane + 16) : lane) & 31;
    const float S = __shfl(e, src, 32);     // row-sum for row (lane), lanes 0-15

    const int p = p0 + m;
    if (lane < 16 && p < P_PRI) {
        const int lab = tcls[n * P_PRI + p];
        const float lse = rmax + __logf(S);
        ce_g[n * P_PRI + p] = lse - row[lab];
    }
}

// ---------------------------------------------------------------------------
// Kernel 3: per-image stats + exact hard-negative mining (bitonic top-k sum).
// One 1024-thread block per image; 64 KB LDS sort buffer.
// ---------------------------------------------------------------------------
__global__ void __launch_bounds__(1024)
mbl_image_loss_kernel(const float* __restrict__ pred_locs,  // (N,P,4)
                      const float* __restrict__ boxes,      // (N,M,4)
                      const float* __restrict__ priors,     // (P,4)
                      const int*   __restrict__ tcls,
                      const int*   __restrict__ obj_g,
                      const float* __restrict__ ce_g,
                      float* __restrict__ img)               // (N,4) stats
{
    const int n   = blockIdx.x;
    const int tid = threadIdx.x;
    const int NT  = 1024;

    __shared__ float buf[SORT_N];   // 64 KB

    // Phase 1: positives: count, ce sum, smooth-L1 sum
    float cnt = 0.0f, cepos = 0.0f, locsum = 0.0f;
    for (int p = tid; p < P_PRI; p += NT) {
        const int tc = tcls[n * P_PRI + p];
        if (tc != 0) {
            cnt   += 1.0f;
            cepos += ce_g[n * P_PRI + p];
            const int m = obj_g[n * P_PRI + p];
            const float x1 = boxes[(n * M_OBJ + m) * 4 + 0];
            const float y1 = boxes[(n * M_OBJ + m) * 4 + 1];
            const float x2 = boxes[(n * M_OBJ + m) * 4 + 2];
            const float y2 = boxes[(n * M_OBJ + m) * 4 + 3];
            const float bcx = (x1 + x2) * 0.5f, bcy = (y1 + y2) * 0.5f;
            const float bw = x2 - x1, bh = y2 - y1;
            const float pcx = priors[p * 4 + 0], pcy = priors[p * 4 + 1];
            const float pw  = priors[p * 4 + 2], ph  = priors[p * 4 + 3];
            const float t0 = (bcx - pcx) / (pw * 0.1f);
            const float t1 = (bcy - pcy) / (ph * 0.1f);
            const float t2 = __logf(bw / pw) * 5.0f;
            const float t3 = __logf(bh / ph) * 5.0f;
            const float* pl = pred_locs + (size_t)(n * P_PRI + p) * 4;
            locsum += smooth_l1(pl[0] - t0) + smooth_l1(pl[1] - t1) +
                      smooth_l1(pl[2] - t2) + smooth_l1(pl[3] - t3);
        }
    }
    buf[tid] = cnt; buf[1024 + tid] = cepos; buf[2048 + tid] = locsum;
    __syncthreads();
    for (int s = NT >> 1; s > 0; s >>= 1) {
        if (tid < s) {
            buf[tid]        += buf[tid + s];
            buf[1024 + tid] += buf[1024 + tid + s];
            buf[2048 + tid] += buf[2048 + tid + s];
        }
        __syncthreads();
    }
    const float nposf = buf[0];
    if (tid == 0) {
        img[n * 4 + 0] = nposf;
        img[n * 4 + 1] = buf[1024];
        img[n * 4 + 2] = buf[2048];
    }
    __syncthreads();

    // Phase 2: fill sort buffer with ce_neg (positives -> 0, pads -> -1)
    for (int i = tid; i < SORT_N; i += NT) {
        float v = -1.0f;
        if (i < P_PRI) {
            const int tc = tcls[n * P_PRI + i];
            v = (tc != 0) ? 0.0f : ce_g[n * P_PRI + i];
        }
        buf[i] = v;
    }
    __syncthreads();

    // Bitonic sort, descending
    for (int k = 2; k <= SORT_N; k <<= 1) {
        for (int j = k >> 1; j > 0; j >>= 1) {
            for (int i = tid; i < SORT_N; i += NT) {
                const int ixj = i ^ j;
                if (ixj > i) {
                    const float a = buf[i], b = buf[ixj];
                    const bool up = ((i & k) == 0);
                    if ((a < b) == up) { buf[i] = b; buf[ixj] = a; }
                }
            }
            __syncthreads();
        }
    }

    // Phase 3: sum top min(3*n_pos, P) hard negatives
    int ktake = 3 * (int)nposf;
    if (ktake > P_PRI) ktake = P_PRI;
    float part = 0.0f;
    for (int i = tid; i < ktake; i += NT) part += buf[i];
    float* red = buf + (SORT_N - 1024);   // tail region (holds spent -1 pads)
    __syncthreads();
    red[tid] = part;
    __syncthreads();
    for (int s = 512; s > 0; s >>= 1) {
        if (tid < s) red[tid] += red[tid + s];
        __syncthreads();
    }
    if (tid == 0) img[n * 4 + 3] = red[0];
}

// ---------------------------------------------------------------------------
// Kernel 4: combine per-image stats into the scalar loss.
// ---------------------------------------------------------------------------
__global__ void mbl_final_kernel(const float* __restrict__ img, float* __restrict__ out)
{
    if (threadIdx.x == 0) {
        float npt = 0.f, cp = 0.f, ls = 0.f, hn = 0.f;
        for (int n = 0; n < N_IMG; ++n) {
            npt += img[n * 4 + 0];
            cp  += img[n * 4 + 1];
            ls  += img[n * 4 + 2];
            hn  += img[n * 4 + 3];
        }
        const float loc_loss  = ls / (npt * 4.0f);
        const float conf_loss = (cp + hn) / npt;
        out[0] = conf_loss + 1.0f * loc_loss;   // ALPHA = 1.0
    }
}

// ---------------------------------------------------------------------------
extern "C" void kernel_launch(void* const* d_in, const int* in_sizes, int n_in,
                              void* d_out, int out_size, void* d_ws, size_t ws_size,
                              hipStream_t stream) {
    const float* predicted_locs   = (const float*)d_in[0];  // N*P*4
    const float* predicted_scores = (const float*)d_in[1];  // N*P*C
    const float* boxes            = (const float*)d_in[2];  // N*M*4
    const int*   labels           = (const int*)  d_in[3];  // N*M
    const float* priors           = (const float*)d_in[4];  // P*4
    float* out = (float*)d_out;

    const size_t np = (size_t)N_IMG * P_PRI;
    float* ov   = (float*)d_ws;          // N*P
    int*   obj  = (int*)(ov + np);       // N*P
    int*   tcls = obj + np;              // N*P
    float* ce   = (float*)(tcls + np);   // N*P
    float* img  = ce + np;               // N*4

    mbl_match_kernel<<<N_IMG, 256, 0, stream>>>(boxes, labels, priors, ov, obj, tcls);

    dim3 grid_ce((P_PRI + 127) / 128, N_IMG);
    mbl_ce_kernel<<<grid_ce, 256, 0, stream>>>(predicted_scores, tcls, ce);

    mbl_image_loss_kernel<<<N_IMG, 1024, 0, stream>>>(predicted_locs, boxes, priors,
                                                      tcls, obj, ce, img);

    mbl_final_kernel<<<1, 32, 0, stream>>>(img, out);
}